// MBCGCN_12223476924780
// MI455X (gfx1250) — compile-verified
//
#include <hip/hip_runtime.h>
#include <hip/hip_bf16.h>

// Problem constants (match reference)
#define U_N   200001
#define I_N   100001
#define NN    (U_N + I_N)      // 300002 nodes
#define DD    64
#define EPB   1000000
#define BCNT  3

typedef __attribute__((ext_vector_type(2))) float v2f;
typedef __attribute__((ext_vector_type(8))) float v8f;

__device__ __forceinline__ void atomAdd(float* p, float v) {
    __hip_atomic_fetch_add(p, v, __ATOMIC_RELAXED, __HIP_MEMORY_SCOPE_AGENT);
}

// ---- degree count (1 thread / edge) ----
__global__ __launch_bounds__(256)
void k_degree(const int* __restrict__ eu, const int* __restrict__ ei,
              float* __restrict__ du, float* __restrict__ di, int n) {
    int e = blockIdx.x * 256 + threadIdx.x;
    if (e >= n) return;
    atomAdd(&du[eu[e]], 1.0f);
    atomAdd(&di[ei[e]], 1.0f);
}

// ---- per-edge normalization ----
__global__ __launch_bounds__(256)
void k_norm(const int* __restrict__ eu, const int* __restrict__ ei,
            const float* __restrict__ du, const float* __restrict__ di,
            float* __restrict__ nrm, int n) {
    int e = blockIdx.x * 256 + threadIdx.x;
    if (e >= n) return;
    float a = du[eu[e]]; a = a < 1.0f ? 1.0f : a;
    float b = di[ei[e]]; b = b < 1.0f ? 1.0f : b;
    nrm[e] = rsqrtf(a * b);
}

// ---- LightGCN propagation: wave (32 lanes) per edge, float2 per lane over D=64 ----
// dst[u]   += nrm * src[U + i]
// dst[U+i] += nrm * src[u]
__global__ __launch_bounds__(256)
void k_scatter(const float* __restrict__ src, float* __restrict__ dst,
               const int* __restrict__ eu, const int* __restrict__ ei,
               const float* __restrict__ nrm, int n) {
    int wid  = (blockIdx.x * 256 + threadIdx.x) >> 5;   // edge index (wave-uniform)
    int lane = threadIdx.x & 31;
    if (wid >= n) return;
    int   u  = eu[wid];
    int   it = ei[wid];
    float w  = nrm[wid];
    const float2* su = (const float2*)(src + (size_t)u * DD);
    const float2* si = (const float2*)(src + ((size_t)U_N + it) * DD);
    float* dstu = dst + (size_t)u * DD;
    float* dsti = dst + ((size_t)U_N + it) * DD;
    float2 vi = si[lane];
    float2 vu = su[lane];
    atomAdd(&dstu[lane * 2 + 0], w * vi.x);
    atomAdd(&dstu[lane * 2 + 1], w * vi.y);
    atomAdd(&dsti[lane * 2 + 0], w * vu.x);
    atomAdd(&dsti[lane * 2 + 1], w * vu.y);
}

// ---- acc += e (float4) ----
__global__ __launch_bounds__(256)
void k_add4(float4* __restrict__ acc, const float4* __restrict__ e, unsigned n4) {
    unsigned i = blockIdx.x * 256 + threadIdx.x;
    if (i >= n4) return;
    float4 a = acc[i], b = e[i];
    a.x += b.x; a.y += b.y; a.z += b.z; a.w += b.w;
    acc[i] = a;
}

// ---- e0 = acc/3 ; sum += e0 (float4) ----
__global__ __launch_bounds__(256)
void k_finish4(const float4* __restrict__ acc, float4* __restrict__ e0,
               float4* __restrict__ sum, unsigned n4) {
    unsigned i = blockIdx.x * 256 + threadIdx.x;
    if (i >= n4) return;
    const float s = 1.0f / 3.0f;
    float4 a = acc[i];
    a.x *= s; a.y *= s; a.z *= s; a.w *= s;
    e0[i] = a;
    float4 t = sum[i];
    t.x += a.x; t.y += a.y; t.z += a.z; t.w += a.w;
    sum[i] = t;
}

// ---- out = A @ W^T, A:(nrows x 64), W:(64 x 64) row-major, via f32 WMMA 16x16x4 ----
// One wave per 16x16 output tile; K looped 0..63 step 4 (16 WMMAs).
__global__ __launch_bounds__(256)
void k_gemm(const float* __restrict__ A, const float* __restrict__ W,
            float* __restrict__ Bout, int nrows) {
    int lane   = threadIdx.x & 31;
    int tile   = blockIdx.x * 8 + (threadIdx.x >> 5);   // wave-uniform
    int ntiles = ((nrows + 15) >> 4) * 4;               // row tiles * (64/16) col tiles
    if (tile >= ntiles) return;                          // whole wave exits: EXEC all-1s for WMMA
    int rowBase = (tile >> 2) * 16;
    int colBase = (tile & 3) * 16;
    int lr = lane & 15;            // M (A) / N (B) within tile
    int kh = (lane >> 4) << 1;     // lane-half selects K offset 0 or 2
    int arow = rowBase + lr;
    if (arow >= nrows) arow = nrows - 1;                 // clamp loads; stores predicated below
    int wrow = colBase + lr;                             // output column -> row of W (out = A W^T)
    const float* ap = A + (size_t)arow * DD + kh;
    const float* wp = W + (size_t)wrow * DD + kh;
    v8f c = {};
#pragma unroll
    for (int k = 0; k < DD; k += 4) {
        v2f a = *(const v2f*)(ap + k);                   // A[M][k+kh], A[M][k+kh+1]
        v2f b = *(const v2f*)(wp + k);                   // W[N][k+kh], W[N][k+kh+1] == B^T
        c = __builtin_amdgcn_wmma_f32_16x16x4_f32(
                false, a, false, b, (short)0, c, false, false);
    }
    // D layout: VGPR v -> row rowBase + v + 8*(lane>=16), col colBase + (lane&15)
    int rOff = rowBase + ((lane >> 4) << 3);
#pragma unroll
    for (int v = 0; v < 8; v++) {
        int r = rOff + v;
        if (r < nrows) Bout[(size_t)r * DD + colBase + lr] = c[v];
    }
}

// ---- BPR loss: thread per (pos r, neg j) pair, 32768 total ----
__global__ __launch_bounds__(256)
void k_loss(const int* __restrict__ x, const float* __restrict__ sum,
            float* __restrict__ lacc) {
    int t = blockIdx.x * 256 + threadIdx.x;   // 0..32767
    int r = t >> 2, j = t & 3;
    const int* row = x + r * 24;              // x[r] is (6,4) int32
    int pu = row[0],             pi = row[1];
    int nu = row[4 + j * 4 + 0], ni = row[4 + j * 4 + 1];
    const float4* puv = (const float4*)(sum + (size_t)pu * DD);
    const float4* piv = (const float4*)(sum + ((size_t)U_N + pi) * DD);
    const float4* nuv = (const float4*)(sum + (size_t)nu * DD);
    const float4* niv = (const float4*)(sum + ((size_t)U_N + ni) * DD);
    float ps = 0.0f, ns = 0.0f;
#pragma unroll 4
    for (int d = 0; d < 16; d++) {
        float4 a = puv[d], b = piv[d];
        ps += a.x * b.x + a.y * b.y + a.z * b.z + a.w * b.w;
        float4 cc = nuv[d], e = niv[d];
        ns += cc.x * e.x + cc.y * e.y + cc.z * e.z + cc.w * e.w;
    }
    float s   = ps - ns;
    float sig = 1.0f / (1.0f + __expf(-s));
    float l   = -__logf(1e-10f + sig);
    __shared__ float red[256];
    red[threadIdx.x] = l;
    __syncthreads();
    for (int off = 128; off > 0; off >>= 1) {
        if (threadIdx.x < off) red[threadIdx.x] += red[threadIdx.x + off];
        __syncthreads();
    }
    if (threadIdx.x == 0) atomAdd(lacc, red[0]);
}

__global__ __launch_bounds__(64)
void k_final(const float* __restrict__ lacc, float* __restrict__ out) {
    if (threadIdx.x == 0 && blockIdx.x == 0)
        out[0] = lacc[0] * (1.0f / 32768.0f);
}

extern "C" void kernel_launch(void* const* d_in, const int* in_sizes, int n_in,
                              void* d_out, int out_size, void* d_ws, size_t ws_size,
                              hipStream_t stream) {
    (void)in_sizes; (void)n_in; (void)out_size; (void)ws_size;
    const int*   x        = (const int*)d_in[0];
    const float* user_emb = (const float*)d_in[1];
    const float* item_emb = (const float*)d_in[2];
    const float* Wu       = (const float*)d_in[3];
    const float* Wi       = (const float*)d_in[4];
    const int*   edges_u  = (const int*)d_in[5];
    const int*   edges_i  = (const int*)d_in[6];
    float*       out      = (float*)d_out;

    const size_t embElems = (size_t)NN * DD;          // 19,200,128 floats
    const size_t embBytes = embElems * sizeof(float); // ~76.8 MB
    char*  ws   = (char*)d_ws;
    float* e0   = (float*)(ws + 0 * embBytes);
    float* e1   = (float*)(ws + 1 * embBytes);
    float* acc  = (float*)(ws + 2 * embBytes);
    float* sum  = (float*)(ws + 3 * embBytes);
    float* du   = (float*)(ws + 4 * embBytes);
    float* di   = du + U_N;
    float* nrm  = di + I_N;
    float* lacc = nrm + EPB;

    // all_emb = concat(user_emb, item_emb); sum = 0
    hipMemcpyAsync(e0, user_emb, (size_t)U_N * DD * sizeof(float),
                   hipMemcpyDeviceToDevice, stream);
    hipMemcpyAsync(e0 + (size_t)U_N * DD, item_emb, (size_t)I_N * DD * sizeof(float),
                   hipMemcpyDeviceToDevice, stream);
    hipMemsetAsync(sum, 0, embBytes, stream);

    const int edgeBlocks     = (EPB + 255) / 256;        // 1 thread/edge
    const int waveEdgeBlocks = (EPB + 7) / 8;            // 1 wave/edge, 8 waves/block
    const unsigned n4        = (unsigned)(embElems / 4);
    const int vec4Blocks     = (int)((n4 + 255) / 256);

    for (int b = 0; b < BCNT; b++) {
        const int* eu = edges_u + (size_t)b * EPB;
        const int* ei = edges_i + (size_t)b * EPB;
        hipMemsetAsync(du, 0, (size_t)(U_N + I_N) * sizeof(float), stream);
        k_degree<<<edgeBlocks, 256, 0, stream>>>(eu, ei, du, di, EPB);
        k_norm  <<<edgeBlocks, 256, 0, stream>>>(eu, ei, du, di, nrm, EPB);

        // acc = e0 (layer-0 term); layer 1 -> e1; acc += e1; layer 2 scatters straight into acc
        hipMemcpyAsync(acc, e0, embBytes, hipMemcpyDeviceToDevice, stream);
        hipMemsetAsync(e1, 0, embBytes, stream);
        k_scatter<<<waveEdgeBlocks, 256, 0, stream>>>(e0, e1, eu, ei, nrm, EPB);
        k_add4   <<<vec4Blocks, 256, 0, stream>>>((float4*)acc, (const float4*)e1, n4);
        k_scatter<<<waveEdgeBlocks, 256, 0, stream>>>(e1, acc, eu, ei, nrm, EPB);
        // e0 = acc/3 ; sum += e0
        k_finish4<<<vec4Blocks, 256, 0, stream>>>((const float4*)acc, (float4*)e0,
                                                  (float4*)sum, n4);

        if (b < BCNT - 1) {
            const float* Wub = Wu + (size_t)b * DD * DD;
            const float* Wib = Wi + (size_t)b * DD * DD;
            int utiles = ((U_N + 15) / 16) * 4;
            int itiles = ((I_N + 15) / 16) * 4;
            k_gemm<<<(utiles + 7) / 8, 256, 0, stream>>>(e0, Wub, e1, U_N);
            k_gemm<<<(itiles + 7) / 8, 256, 0, stream>>>(e0 + (size_t)U_N * DD, Wib,
                                                         e1 + (size_t)U_N * DD, I_N);
            float* t = e0; e0 = e1; e1 = t;   // transformed embeddings become current
        }
    }

    hipMemsetAsync(lacc, 0, sizeof(float), stream);
    k_loss <<<32768 / 256, 256, 0, stream>>>(x, sum, lacc);
    k_final<<<1, 64, 0, stream>>>(lacc, out);
}